// DCE_25271587569966
// MI455X (gfx1250) — compile-verified
//
#include <hip/hip_runtime.h>
#include <hip/hip_bf16.h>
#include <math.h>

typedef __attribute__((ext_vector_type(2))) float v2f;
typedef __attribute__((ext_vector_type(8))) float v8f;
typedef int v4i __attribute__((vector_size(16)));   // matches builtin param

#define DDIM    2048
#define NROWS   40000            // C*K = 10000*4
#define TILE    16               // rows per wave / WMMA tile
#define NTILES  (NROWS / TILE)   // 2500
#define WPB     8                // waves per block
#define CHUNK   32               // dims staged per round
#define NCHUNKS (DDIM / CHUNK)   // 64
#define PSTR    36               // padded LDS row stride (floats); 36m mod 64
                                 // is a permutation of multiples of 4 -> all
                                 // 64 banks hit exactly once per access

#define AS1 __attribute__((address_space(1)))
#define AS3 __attribute__((address_space(3)))

#if defined(__AMDGCN__) && \
    __has_builtin(__builtin_amdgcn_global_load_async_to_lds_b128) && \
    __has_builtin(__builtin_amdgcn_s_wait_asynccnt)
#define USE_ASYNC 1
#else
#define USE_ASYNC 0
#endif

// ---------------------------------------------------------------------------
// Kernel 1: per-wave 16-row tile. dist2[row] = sum_d (p[row,d]-f[d])^2 as the
// diagonal of (P-F)(P-F)^T via v_wmma_f32_16x16x4_f32, K=4 dims per WMMA,
// full f32. Prototype stream staged global->LDS with double-buffered async
// copies (ASYNCcnt) so HBM traffic overlaps the WMMA pipeline.
// ---------------------------------------------------------------------------
__global__ __launch_bounds__(256) void dce_dist_kernel(
    const float* __restrict__ feature,
    const float* __restrict__ prototypes,
    float* __restrict__ dist2)
{
    __shared__ float sF[DDIM];                    // 8 KB feature
    __shared__ float sP[2][WPB][TILE * PSTR];     // 2 x 8 x 2304 B staging

    const int tid  = threadIdx.x;
    const int wave = tid >> 5;
    const int lane = tid & 31;

    for (int i = tid; i < DDIM; i += 256) sF[i] = feature[i];
    __syncthreads();

    const int tile = blockIdx.x * WPB + wave;
    if (tile >= NTILES) return;                   // wave-uniform exit

    const float* rowBase = prototypes + (size_t)tile * TILE * DDIM;

    const int half = lane >> 4;                   // 0: K 0..1, 1: K 2..3
    const int m    = lane & 15;                   // row within tile
    const int rq   = lane >> 3;                   // staging row quarter 0..3
    const int cc   = (lane & 7) * 4;              // staging column (floats)

#if USE_ASYNC
    // 4 async b128 copies stage one 16x32 chunk (each copies 4 rows x 128 B,
    // coalesced per 8-lane group) straight into LDS, no VGPR landing zone.
    auto issue_stage = [&](int kb, float* dstBase) {
#pragma unroll
        for (int i = 0; i < 4; ++i) {
            const int r = i * 4 + rq;
            __builtin_amdgcn_global_load_async_to_lds_b128(
                (AS1 v4i*)(rowBase + (size_t)r * DDIM + kb + cc),
                (AS3 v4i*)(dstBase + r * PSTR + cc),
                0, 0);
        }
    };
    issue_stage(0, &sP[0][wave][0]);              // prologue: chunk 0 in flight
#endif

    v8f acc = {};

    for (int kbi = 0; kbi < NCHUNKS; ++kbi) {
        const int kb = kbi * CHUNK;
        float* bufP = &sP[kbi & 1][wave][0];

#if USE_ASYNC
        if (kbi + 1 < NCHUNKS) {
            issue_stage(kb + CHUNK, &sP[(kbi + 1) & 1][wave][0]);
            __builtin_amdgcn_s_wait_asynccnt(4);  // current chunk landed
        } else {
            __builtin_amdgcn_s_wait_asynccnt(0);
        }
        __asm__ volatile("" ::: "memory");        // keep ds reads below wait
#else
        // fallback: sync stage through VGPRs
#pragma unroll
        for (int i = 0; i < 4; ++i) {
            const int r = i * 4 + rq;
            const float4 v =
                *(const float4*)(rowBase + (size_t)r * DDIM + kb + cc);
            *(float4*)&bufP[r * PSTR + cc] = v;
        }
#endif

        // 8 WMMA steps over the staged chunk (K advances by 4). The f32 16x4
        // A layout equals the 4x16 B layout read as A^T, so feeding the diff
        // registers as both operands accumulates the Gram matrix.
#pragma unroll
        for (int ks = 0; ks < CHUNK; ks += 4) {
            const int kk = ks + half * 2;
            const v2f p = *(const v2f*)&bufP[m * PSTR + kk];   // ds_load_b64
            const v2f f = *(const v2f*)&sF[kb + kk];           // broadcast
            v2f a;
            a.x = p.x - f.x;
            a.y = p.y - f.y;
            acc = __builtin_amdgcn_wmma_f32_16x16x4_f32(
                false, a, false, a, (short)0, acc, false, false);
        }
    }

    // diagonal (m,m): lane m / vgpr m (m<8); lane m+16 / vgpr m-8 (m>=8)
    if (lane < 8 || lane >= 24) {
        const int j = (lane < 8) ? lane : (lane - 24);
        float v;
        switch (j) {
            case 0: v = acc[0]; break; case 1: v = acc[1]; break;
            case 2: v = acc[2]; break; case 3: v = acc[3]; break;
            case 4: v = acc[4]; break; case 5: v = acc[5]; break;
            case 6: v = acc[6]; break; default: v = acc[7]; break;
        }
        const int row = tile * TILE + ((lane < 8) ? lane : (lane - 16));
        dist2[row] = v;
    }
}

// ---------------------------------------------------------------------------
// Kernel 2: single block. Stable logsumexp over logits = -dist2, then
// probability = 4*log_one + sum_k dist2[label*4+k].
// ---------------------------------------------------------------------------
__global__ __launch_bounds__(1024) void dce_lse_kernel(
    const float* __restrict__ dist2,
    const int* __restrict__ label,
    float* __restrict__ out)
{
    __shared__ float red[1024];
    const int tid = threadIdx.x;

    float mn = 3.402823466e+38f;
    for (int i = tid; i < NROWS; i += 1024) mn = fminf(mn, dist2[i]);
    red[tid] = mn;
    __syncthreads();
    for (int s = 512; s > 0; s >>= 1) {
        if (tid < s) red[tid] = fminf(red[tid], red[tid + s]);
        __syncthreads();
    }
    mn = red[0];
    __syncthreads();

    float sm = 0.0f;
    for (int i = tid; i < NROWS; i += 1024) sm += expf(mn - dist2[i]);
    red[tid] = sm;
    __syncthreads();
    for (int s = 512; s > 0; s >>= 1) {
        if (tid < s) red[tid] += red[tid + s];
        __syncthreads();
    }

    if (tid == 0) {
        const int L = label[0];
        const float log_one = -mn + logf(red[0]);
        const float lbl = dist2[L * 4 + 0] + dist2[L * 4 + 1] +
                          dist2[L * 4 + 2] + dist2[L * 4 + 3];
        out[0] = 4.0f * log_one + lbl;
    }
}

extern "C" void kernel_launch(void* const* d_in, const int* in_sizes, int n_in,
                              void* d_out, int out_size, void* d_ws, size_t ws_size,
                              hipStream_t stream) {
    (void)in_sizes; (void)n_in; (void)out_size; (void)ws_size;
    const float* feature    = (const float*)d_in[0];
    const int*   label      = (const int*)d_in[1];
    const float* prototypes = (const float*)d_in[2];
    float*       out        = (float*)d_out;
    float*       ws         = (float*)d_ws;      // 40000 floats (160 KB)

    const int grid1 = (NTILES + WPB - 1) / WPB;  // 313 blocks x 8 waves
    dce_dist_kernel<<<grid1, 256, 0, stream>>>(feature, prototypes, ws);
    dce_lse_kernel<<<1, 1024, 0, stream>>>(ws, label, out);
}